// RoIHeads_39109972197403
// MI455X (gfx1250) — compile-verified
//
#include <hip/hip_runtime.h>
#include <cstdint>
#include <cstddef>

// ---------------------------------------------------------------------------
// RoIHeads post-process for MI455X (gfx1250, wave32).
//   K1: softmax + box decode/clip/mask -> padded [B*CM1,1024] planes in d_ws
//   K2: per-(b,c): TDM (tensor_load_to_lds) stages the 16KB box plane,
//       per-lane async DMA stages scores; bitonic sort; greedy NMS; top-K.
// ---------------------------------------------------------------------------

namespace {
constexpr int kB    = 8;
constexpr int kN    = 1000;
constexpr int kC    = 21;
constexpr int kCM1  = 20;    // foreground classes
constexpr int kK    = 100;   // detections per class
constexpr int kNPad = 1024;  // padded proposal count (power of two for bitonic)
constexpr int kThr  = 256;

constexpr float kScoreThresh = 0.05f;
constexpr float kNmsThresh   = 0.5f;
constexpr float kMinSize     = 1.0f;
constexpr float kBboxClip    = 4.135166556742356f;  // log(1000/16)
}  // namespace

typedef __attribute__((ext_vector_type(4))) unsigned uint32x4;
typedef __attribute__((ext_vector_type(8))) unsigned uint32x8;

// ---- gfx1250 async DMA to LDS (ASYNCcnt path) ------------------------------

__device__ __forceinline__ unsigned to_lds_addr(const void* p) {
  // generic -> LDS address space; low 32 bits are the LDS byte address.
  return (unsigned)(unsigned long long)(
      (__attribute__((address_space(3))) const void*)p);
}

__device__ __forceinline__ void async_copy_b32(void* lds_dst, const void* gsrc) {
  asm volatile("global_load_async_to_lds_b32 %0, %1, off"
               :: "v"(to_lds_addr(lds_dst)),
                  "v"((unsigned long long)(uintptr_t)gsrc)
               : "memory");
}

__device__ __forceinline__ void wait_async_zero() {
  asm volatile("s_wait_asynccnt 0" ::: "memory");
}

// ---- gfx1250 Tensor Data Mover: one-shot contiguous tile into LDS ----------
// Copies n_dwords (<= 65535) contiguous dwords from gsrc to LDS at lds_byte.
// Wave-level op (EXEC ignored); call from one wave only; tracked by TENSORcnt.
__device__ __forceinline__ void tdm_load_tile_to_lds(unsigned lds_byte,
                                                     const void* gsrc,
                                                     unsigned n_dwords) {
  const unsigned long long ga = (unsigned long long)(uintptr_t)gsrc;
  unsigned ga_lo = __builtin_amdgcn_readfirstlane((unsigned)ga);
  unsigned ga_hi = __builtin_amdgcn_readfirstlane((unsigned)(ga >> 32));
  unsigned lds   = __builtin_amdgcn_readfirstlane(lds_byte);
  unsigned nd    = __builtin_amdgcn_readfirstlane(n_dwords);

  uint32x4 g0;
  g0[0] = 1u;                                   // count=1 (valid), user mode
  g0[1] = lds;                                  // lds_addr (bytes)
  g0[2] = ga_lo;                                // global_addr[31:0]
  g0[3] = (ga_hi & 0x1FFFFFFu) | 0x80000000u;   // global_addr[56:32] | type=2

  uint32x8 g1;
  g1[0] = 0x00020000u;                          // wg_mask=0, data_size=2 (4B)
  g1[1] = (nd & 0xFFFFu) << 16;                 // tensor_dim0[15:0]
  g1[2] = ((nd >> 16) & 0xFFFFu) | (1u << 16);  // tensor_dim0[31:16], tensor_dim1=1
  g1[3] = (nd & 0xFFFFu) << 16;                 // tensor_dim1 hi=0, tile_dim0=nd
  g1[4] = 0x00000001u;                          // tile_dim1=1, tile_dim2=0
  g1[5] = nd;                                   // tensor_dim0_stride[31:0]
  g1[6] = 0u;                                   // stride hi / dim1_stride lo
  g1[7] = 0u;

  asm volatile("tensor_load_to_lds %0, %1" :: "s"(g0), "s"(g1) : "memory");
}

// ---------------------------------------------------------------------------
// Kernel 1: softmax + decode + clip + validity mask.
// ---------------------------------------------------------------------------
__global__ __launch_bounds__(kThr) void roi_decode_kernel(
    const float* __restrict__ logits,   // [B,N,C]
    const float* __restrict__ deltas,   // [B,N,C*4]
    const float* __restrict__ props,    // [B,N,4]
    const int*   __restrict__ img_hw,   // [B,2] (H,W)
    float* __restrict__ scores_ws,      // [B*CM1, kNPad]
    float* __restrict__ boxes_ws)       // [B*CM1, kNPad, 4]
{
  const int t = blockIdx.x * kThr + threadIdx.x;
  const int b = t >> 10;            // / kNPad
  const int n = t & (kNPad - 1);
  if (b >= kB) return;

  if (n >= kN) {                    // padding entries: sink to the end
    const float4 z = make_float4(0.f, 0.f, 0.f, 0.f);
    for (int c = 0; c < kCM1; ++c) {
      const int row = b * kCM1 + c;
      scores_ws[row * kNPad + n] = -1.0f;
      reinterpret_cast<float4*>(boxes_ws)[row * kNPad + n] = z;
    }
    return;
  }

  const size_t bn = (size_t)b * kN + n;
  const float* lg = logits + bn * kC;
  const float* dl = deltas + bn * (size_t)(kC * 4);
  __builtin_prefetch(dl, 0, 1);     // global_prefetch_b8

  // Softmax over all 21 classes (background included in the denominator).
  float m = lg[0];
#pragma unroll
  for (int c = 1; c < kC; ++c) m = fmaxf(m, lg[c]);
  float e[kC];
  float sum = 0.0f;
#pragma unroll
  for (int c = 0; c < kC; ++c) { e[c] = expf(lg[c] - m); sum += e[c]; }
  const float inv_sum = 1.0f / sum;

  const float px1 = props[bn * 4 + 0], py1 = props[bn * 4 + 1];
  const float px2 = props[bn * 4 + 2], py2 = props[bn * 4 + 3];
  const float w  = px2 - px1, h = py2 - py1;
  const float cx = px1 + 0.5f * w, cy = py1 + 0.5f * h;
  const float Hf = (float)img_hw[b * 2 + 0];
  const float Wf = (float)img_hw[b * 2 + 1];

  for (int c = 1; c < kC; ++c) {
    const float prob = e[c] * inv_sum;
    const float* d = dl + c * 4;
    const float dx = d[0] / 10.0f;
    const float dy = d[1] / 10.0f;
    const float dw = fminf(d[2] / 5.0f, kBboxClip);
    const float dh = fminf(d[3] / 5.0f, kBboxClip);
    const float pcx = dx * w + cx;
    const float pcy = dy * h + cy;
    const float pw  = expf(dw) * w;
    const float ph  = expf(dh) * h;
    const float x1 = fminf(fmaxf(pcx - 0.5f * pw, 0.0f), Wf);
    const float y1 = fminf(fmaxf(pcy - 0.5f * ph, 0.0f), Hf);
    const float x2 = fminf(fmaxf(pcx + 0.5f * pw, 0.0f), Wf);
    const float y2 = fminf(fmaxf(pcy + 0.5f * ph, 0.0f), Hf);
    const bool valid = (prob >= kScoreThresh) &&
                       ((x2 - x1) >= kMinSize) && ((y2 - y1) >= kMinSize);
    const int row = b * kCM1 + (c - 1);
    scores_ws[row * kNPad + n] = valid ? prob : -1.0f;
    reinterpret_cast<float4*>(boxes_ws)[row * kNPad + n] =
        make_float4(x1, y1, x2, y2);
  }
}

// ---------------------------------------------------------------------------
// Kernel 2: per-(b,c) sort + greedy NMS + top-K emit. One block per task.
// ---------------------------------------------------------------------------
__global__ __launch_bounds__(kThr) void roi_nms_kernel(
    const float* __restrict__ scores_ws,  // [B*CM1, kNPad]
    const float* __restrict__ boxes_ws,   // [B*CM1, kNPad, 4]
    float* __restrict__ out)              // [B*CM1, kK, 5]
{
  const int bc  = blockIdx.x;   // b*kCM1 + (c-1)
  const int tid = threadIdx.x;

  __shared__ float4             rawbox[kNPad];   // 16 KB  (TDM destination)
  __shared__ float              rawsco[kNPad];   //  4 KB  (async-DMA dest)
  __shared__ unsigned long long key[kNPad];      //  8 KB
  __shared__ float4             sbox[kNPad];     // 16 KB
  __shared__ float              ss[kNPad];       //  4 KB
  __shared__ int                keep[kNPad];     //  4 KB
  __shared__ int                tsum[kThr];      //  1 KB

  const float* gbox = boxes_ws  + (size_t)bc * kNPad * 4;
  const float* gsco = scores_ws + (size_t)bc * kNPad;

  // ---- Stage the box plane (16KB, 4096 dwords) with ONE TDM descriptor. ---
  if (tid == 0) {
    tdm_load_tile_to_lds(to_lds_addr(&rawbox[0]), gbox, kNPad * 4);
  }
  // ---- Stage scores via the per-lane async-DMA path (ASYNCcnt). -----------
  for (int j = tid; j < kNPad; j += kThr) {
    async_copy_b32(&rawsco[j], gsco + j);
  }
  wait_async_zero();                              // s_wait_asynccnt 0
  if (tid == 0) __builtin_amdgcn_s_wait_tensorcnt(0);
  __syncthreads();

  // ---- Build descending-orderable keys: (score bits || inverted index). ---
  for (int j = tid; j < kNPad; j += kThr) {
    const unsigned u   = __float_as_uint(rawsco[j]);
    const unsigned ord = (u & 0x80000000u) ? ~u : (u | 0x80000000u);
    key[j] = ((unsigned long long)ord << 32) |
             (unsigned)(kNPad - 1 - j);   // ties: smaller index first
  }
  __syncthreads();

  // ---- Bitonic sort, descending (matches jnp.argsort(-s) stability). -----
  for (int k = 2; k <= kNPad; k <<= 1) {
    for (int j = k >> 1; j > 0; j >>= 1) {
      for (int i = tid; i < kNPad; i += kThr) {
        const int l = i ^ j;
        if (l > i) {
          const bool up = ((i & k) == 0);
          const unsigned long long a = key[i];
          const unsigned long long b = key[l];
          if ((a < b) == up) { key[i] = b; key[l] = a; }
        }
      }
      __syncthreads();
    }
  }

  // ---- Gather boxes/scores in sorted order; init keep mask. ---------------
  for (int s = tid; s < kNPad; s += kThr) {
    const int idx = kNPad - 1 - (int)(key[s] & 0xFFFFFFFFu);
    const float sc = rawsco[idx];
    sbox[s] = rawbox[idx];
    ss[s]   = sc;
    keep[s] = (sc >= 0.0f) ? 1 : 0;
  }
  __syncthreads();

  // ---- Greedy NMS: identical semantics to the reference scan. -------------
  for (int i = 0; i < kNPad; ++i) {
    if (ss[i] < 0.0f) break;     // sorted: everything after is invalid
    if (!keep[i]) continue;      // uniform branch; no LDS writes on this path
    const float4 bi = sbox[i];
    const float area_i = (bi.z - bi.x) * (bi.w - bi.y);
    for (int j = i + 1 + tid; j < kNPad; j += kThr) {
      if (!keep[j]) continue;
      const float4 bj = sbox[j];
      const float xx1 = fmaxf(bi.x, bj.x);
      const float yy1 = fmaxf(bi.y, bj.y);
      const float xx2 = fminf(bi.z, bj.z);
      const float yy2 = fminf(bi.w, bj.w);
      const float iw = fmaxf(xx2 - xx1, 0.0f);
      const float ih = fmaxf(yy2 - yy1, 0.0f);
      const float inter  = iw * ih;
      const float area_j = (bj.z - bj.x) * (bj.w - bj.y);
      const float iou = inter / (area_i + area_j - inter + 1e-9f);
      if (iou > kNmsThresh) keep[j] = 0;
    }
    __syncthreads();
  }

  // ---- Compact kept entries (exclusive scan over blocked counts). ---------
  const int j0 = tid * (kNPad / kThr);        // 4 consecutive entries/thread
  int cnt = 0;
#pragma unroll
  for (int j = 0; j < kNPad / kThr; ++j) cnt += keep[j0 + j];
  tsum[tid] = cnt;
  __syncthreads();
  for (int off = 1; off < kThr; off <<= 1) {  // Hillis–Steele inclusive scan
    const int v = (tid >= off) ? tsum[tid - off] : 0;
    __syncthreads();
    tsum[tid] += v;
    __syncthreads();
  }
  const int base  = tsum[tid] - cnt;          // exclusive prefix
  const int total = tsum[kThr - 1];

  // ---- Emit [kK,5]: kept entries in score order, zero-filled tail. --------
  float* obase = out + (size_t)bc * kK * 5;
  const int zstart = (total < kK) ? total : kK;
  for (int s = zstart + tid; s < kK; s += kThr) {
#pragma unroll
    for (int q = 0; q < 5; ++q) obase[s * 5 + q] = 0.0f;
  }
  int r = base;
#pragma unroll
  for (int j = 0; j < kNPad / kThr; ++j) {
    const int idx = j0 + j;
    if (keep[idx]) {
      if (r < kK) {
        const float4 bx = sbox[idx];
        obase[r * 5 + 0] = bx.x;
        obase[r * 5 + 1] = bx.y;
        obase[r * 5 + 2] = bx.z;
        obase[r * 5 + 3] = bx.w;
        obase[r * 5 + 4] = ss[idx];   // kept => >= 0, matches max(vals, 0)
      }
      ++r;
    }
  }
}

// ---------------------------------------------------------------------------
extern "C" void kernel_launch(void* const* d_in, const int* in_sizes, int n_in,
                              void* d_out, int out_size, void* d_ws,
                              size_t ws_size, hipStream_t stream) {
  (void)in_sizes; (void)n_in; (void)out_size; (void)ws_size;

  const float* logits = (const float*)d_in[0];  // [B,N,C]
  const float* deltas = (const float*)d_in[1];  // [B,N,C*4]
  const float* props  = (const float*)d_in[2];  // [B,N,4]
  const int*   img_hw = (const int*)d_in[3];    // [B,2]

  // Workspace layout (floats): scores [B*CM1*kNPad], then boxes [B*CM1*kNPad*4]
  float* scores_ws = (float*)d_ws;
  float* boxes_ws  = scores_ws + (size_t)kB * kCM1 * kNPad;

  const int grid1 = (kB * kNPad) / kThr;     // 32 blocks
  roi_decode_kernel<<<grid1, kThr, 0, stream>>>(logits, deltas, props, img_hw,
                                                scores_ws, boxes_ws);

  const int grid2 = kB * kCM1;               // 160 blocks, one per (b,c)
  roi_nms_kernel<<<grid2, kThr, 0, stream>>>(scores_ws, boxes_ws,
                                             (float*)d_out);
}